// GAT_38585986187618
// MI455X (gfx1250) — compile-verified
//
#include <hip/hip_runtime.h>
#include <hip/hip_bf16.h>
#include <math.h>

#define NN   50000
#define EE   800000
#define ET   (EE + NN)   // edges + self loops
#define FD   128         // H*C
#define NH   8
#define NC   16
#define NG   64
#define NCLS 10
#define NEG_SLOPE 0.2f

typedef __attribute__((ext_vector_type(2))) float v2f;
typedef __attribute__((ext_vector_type(8))) float v8f;

// ---------------------------------------------------------------- utilities
__global__ void gat_fill_kernel(float* p, float v, int n) {
    int t = blockIdx.x * blockDim.x + threadIdx.x;
    if (t < n) p[t] = v;
}

__device__ __forceinline__ void atomic_max_float(float* addr, float v) {
    // valid total order trick: positive floats compare as ints,
    // negative floats compare reversed as uints. init value = -inf bits.
    if (v >= 0.0f) atomicMax((int*)addr, __float_as_int(v));
    else           atomicMin((unsigned int*)addr, __float_as_uint(v));
}

__device__ __forceinline__ void edge_sd(const int* __restrict__ ei, int e,
                                        int& s, int& d) {
    if (e < EE) { s = ei[e]; d = ei[EE + e]; }
    else        { s = e - EE; d = e - EE; }   // self loop
}

// ------------------------------------------------- fp32 WMMA GEMM  H = X @ W
// one wave -> one 16x16 tile of H, K swept 0..127 in steps of 4 via
// V_WMMA_F32_16X16X4_F32. W column-tile (128x16, 8KB) staged in LDS.
__global__ __launch_bounds__(32)
void gat_gemm_wmma(const float* __restrict__ X, const float* __restrict__ W,
                   float* __restrict__ Hout) {
    __shared__ float Wt[FD * 16];          // [k][cc]
    const int lane = threadIdx.x;          // wave32
    const int m0 = blockIdx.x * 16;
    const int n0 = blockIdx.y * 16;

    for (int i = lane; i < FD * 16; i += 32) {
        int k = i >> 4, cc = i & 15;
        Wt[i] = W[k * FD + n0 + cc];
    }
    __syncthreads();

    const int mrow  = m0 + (lane & 15);
    const int khalf = (lane >> 4) * 2;     // lanes 0-15: K+0/K+1, 16-31: K+2/K+3
    const float* __restrict__ xrow = X + (size_t)mrow * FD;

    v8f acc = {};
#pragma unroll
    for (int k = 0; k < FD; k += 4) {
        v2f a, b;
        a.x = xrow[k + khalf];
        a.y = xrow[k + khalf + 1];
        b.x = Wt[(k + khalf)     * 16 + (lane & 15)];
        b.y = Wt[(k + khalf + 1) * 16 + (lane & 15)];
        acc = __builtin_amdgcn_wmma_f32_16x16x4_f32(
                  false, a, false, b, (short)0, acc, false, false);
    }

    const int rbase = m0 + ((lane >> 4) << 3);
    const int col   = n0 + (lane & 15);
#pragma unroll
    for (int r = 0; r < 8; ++r)
        Hout[(size_t)(rbase + r) * FD + col] = acc[r];
}

// ---------------------------------------------- attention coefficient dots
__global__ void gat_attn_coef(const float* __restrict__ H,
                              const float* __restrict__ a_src,
                              const float* __restrict__ a_dst,
                              float* __restrict__ asrc,
                              float* __restrict__ adst) {
    int t = blockIdx.x * blockDim.x + threadIdx.x;   // N*8 threads
    if (t >= NN * NH) return;
    int node = t >> 3, hd = t & 7;
    const float* hp = H + (size_t)node * FD + hd * NC;
    float s1 = 0.f, s2 = 0.f;
#pragma unroll
    for (int c = 0; c < NC; ++c) {
        float v = hp[c];
        s1 += v * a_src[hd * NC + c];
        s2 += v * a_dst[hd * NC + c];
    }
    asrc[t] = s1;
    adst[t] = s2;
}

// ------------------------------------------------------------- edge pass 1
__global__ void gat_edge_max(const int* __restrict__ ei,
                             const float* __restrict__ asrc,
                             const float* __restrict__ adst,
                             float* __restrict__ m) {
    int e = blockIdx.x * blockDim.x + threadIdx.x;
    if (e >= ET) return;
    int s, d; edge_sd(ei, e, s, d);
#pragma unroll
    for (int h = 0; h < NH; ++h) {
        float ev = asrc[s * NH + h] + adst[d * NH + h];
        ev = ev > 0.f ? ev : NEG_SLOPE * ev;
        atomic_max_float(&m[d * NH + h], ev);
    }
}

// ------------------------------------------------------------- edge pass 2
__global__ void gat_edge_exp(const int* __restrict__ ei,
                             const float* __restrict__ asrc,
                             const float* __restrict__ adst,
                             const float* __restrict__ m,
                             float* __restrict__ exbuf,
                             float* __restrict__ sbuf) {
    int e = blockIdx.x * blockDim.x + threadIdx.x;
    if (e >= ET) return;
    int s, d; edge_sd(ei, e, s, d);
#pragma unroll
    for (int h = 0; h < NH; ++h) {
        float ev = asrc[s * NH + h] + adst[d * NH + h];
        ev = ev > 0.f ? ev : NEG_SLOPE * ev;
        float ex = __expf(ev - m[d * NH + h]);
        exbuf[(size_t)e * NH + h] = ex;
        atomicAdd(&sbuf[d * NH + h], ex);
    }
}

// ------------------------------------------- edge pass 3: weighted scatter
// 256 threads = 2 edges x 128 channels (coalesced gather + L2 atomics)
__global__ __launch_bounds__(256)
void gat_edge_aggr(const int* __restrict__ ei,
                   const float* __restrict__ H,
                   const float* __restrict__ exbuf,
                   const float* __restrict__ sbuf,
                   float* __restrict__ out) {
    int e = blockIdx.x * 2 + (threadIdx.x >> 7);
    int j = threadIdx.x & 127;
    if (e >= ET) return;
    int s, d; edge_sd(ei, e, s, d);
    int hd = j >> 4;
    float alpha = exbuf[(size_t)e * NH + hd] / sbuf[d * NH + hd];
    atomicAdd(&out[(size_t)d * FD + j], H[(size_t)s * FD + j] * alpha);
}

// --------------------------------------------------------- bias + ELU
__global__ void gat_bias_elu(float* __restrict__ out,
                             const float* __restrict__ b) {
    int t = blockIdx.x * blockDim.x + threadIdx.x;
    if (t >= NN * FD) return;
    float v = out[t] + b[t & 127];
    out[t] = v > 0.f ? v : expm1f(v);
}

// --------------------------------------------------------- mean pool
__global__ void gat_pool_sum(const float* __restrict__ act,
                             const int* __restrict__ batch,
                             float* __restrict__ psum) {
    int t = blockIdx.x * blockDim.x + threadIdx.x;
    if (t >= NN * FD) return;
    int node = t >> 7, j = t & 127;
    atomicAdd(&psum[(size_t)batch[node] * FD + j], act[t]);
}

__global__ void gat_pool_cnt(const int* __restrict__ batch,
                             float* __restrict__ pcnt) {
    int t = blockIdx.x * blockDim.x + threadIdx.x;
    if (t >= NN) return;
    atomicAdd(&pcnt[batch[t]], 1.0f);
}

// --------------------------------------------------------- MLP head
__global__ __launch_bounds__(64)
void gat_head(const float* __restrict__ psum, const float* __restrict__ pcnt,
              const float* __restrict__ l1W, const float* __restrict__ l1b,
              const float* __restrict__ l2W, const float* __restrict__ l2b,
              float* __restrict__ out) {
    int g = threadIdx.x;
    if (g >= NG) return;
    float inv = 1.0f / fmaxf(pcnt[g], 1.0f);
    float z[NC];
#pragma unroll
    for (int k = 0; k < NC; ++k) {
        float a = l1b[k];
        for (int j = 0; j < FD; ++j)
            a += (psum[(size_t)g * FD + j] * inv) * l1W[j * NC + k];
        z[k] = a > 0.f ? a : expm1f(a);
    }
    float lg[NCLS]; float mx = -1e30f;
#pragma unroll
    for (int c = 0; c < NCLS; ++c) {
        float a = l2b[c];
#pragma unroll
        for (int k = 0; k < NC; ++k) a += z[k] * l2W[k * NCLS + c];
        lg[c] = a; mx = fmaxf(mx, a);
    }
    float se = 0.f;
#pragma unroll
    for (int c = 0; c < NCLS; ++c) se += __expf(lg[c] - mx);
    float lse = mx + __logf(se);
#pragma unroll
    for (int c = 0; c < NCLS; ++c) {
        out[g * NCLS + c]             = lg[c] - lse;   // log_softmax
        out[NG * NCLS + g * NCLS + c] = lg[c];         // logits
    }
}

// ------------------------------------------------------------------ driver
static void run_gat_layer(const float* x_in, const float* W, const float* a_s,
                          const float* a_d, const float* b, const int* ei,
                          float* bufH, float* bufO, float* asrc, float* adst,
                          float* mbuf, float* sbuf, float* exbuf,
                          hipStream_t stream) {
    dim3 ggrid(NN / 16, FD / 16);
    gat_gemm_wmma<<<ggrid, 32, 0, stream>>>(x_in, W, bufH);
    gat_attn_coef<<<(NN * NH + 255) / 256, 256, 0, stream>>>(bufH, a_s, a_d, asrc, adst);
    gat_fill_kernel<<<(NN * NH + 255) / 256, 256, 0, stream>>>(mbuf, -INFINITY, NN * NH);
    gat_fill_kernel<<<(NN * NH + 255) / 256, 256, 0, stream>>>(sbuf, 0.0f, NN * NH);
    gat_fill_kernel<<<(NN * FD + 255) / 256, 256, 0, stream>>>(bufO, 0.0f, NN * FD);
    gat_edge_max <<<(ET + 255) / 256, 256, 0, stream>>>(ei, asrc, adst, mbuf);
    gat_edge_exp <<<(ET + 255) / 256, 256, 0, stream>>>(ei, asrc, adst, mbuf, exbuf, sbuf);
    gat_edge_aggr<<<(ET + 1) / 2, 256, 0, stream>>>(ei, bufH, exbuf, sbuf, bufO);
    gat_bias_elu <<<(NN * FD + 255) / 256, 256, 0, stream>>>(bufO, b);
}

extern "C" void kernel_launch(void* const* d_in, const int* in_sizes, int n_in,
                              void* d_out, int out_size, void* d_ws, size_t ws_size,
                              hipStream_t stream) {
    const float* x      = (const float*)d_in[0];
    const int*   ei     = (const int*)  d_in[1];
    const int*   batch  = (const int*)  d_in[2];
    const float* W1     = (const float*)d_in[3];
    const float* a_src1 = (const float*)d_in[4];
    const float* a_dst1 = (const float*)d_in[5];
    const float* b1     = (const float*)d_in[6];
    const float* W2     = (const float*)d_in[7];
    const float* a_src2 = (const float*)d_in[8];
    const float* a_dst2 = (const float*)d_in[9];
    const float* b2     = (const float*)d_in[10];
    const float* l1W    = (const float*)d_in[11];
    const float* l1b    = (const float*)d_in[12];
    const float* l2W    = (const float*)d_in[13];
    const float* l2b    = (const float*)d_in[14];
    float* out = (float*)d_out;

    float* ws    = (float*)d_ws;
    float* bufH  = ws;                    // N*128
    float* bufO  = bufH  + (size_t)NN * FD;   // N*128
    float* asrc  = bufO  + (size_t)NN * FD;   // N*8
    float* adst  = asrc  + (size_t)NN * NH;
    float* mbuf  = adst  + (size_t)NN * NH;
    float* sbuf  = mbuf  + (size_t)NN * NH;
    float* exbuf = sbuf  + (size_t)NN * NH;   // ET*8
    float* psum  = exbuf + (size_t)ET * NH;   // G*128
    float* pcnt  = psum  + (size_t)NG * FD;   // G

    // layer 1: x -> bufO
    run_gat_layer(x,    W1, a_src1, a_dst1, b1, ei,
                  bufH, bufO, asrc, adst, mbuf, sbuf, exbuf, stream);
    // layer 2: bufO -> bufO (bufO free after its GEMM reads it)
    run_gat_layer(bufO, W2, a_src2, a_dst2, b2, ei,
                  bufH, bufO, asrc, adst, mbuf, sbuf, exbuf, stream);

    // global mean pool + head
    gat_fill_kernel<<<(NG * FD + 255) / 256, 256, 0, stream>>>(psum, 0.0f, NG * FD);
    gat_fill_kernel<<<1, 64, 0, stream>>>(pcnt, 0.0f, NG);
    gat_pool_sum<<<(NN * FD + 255) / 256, 256, 0, stream>>>(bufO, batch, psum);
    gat_pool_cnt<<<(NN + 255) / 256, 256, 0, stream>>>(batch, pcnt);
    gat_head<<<1, 64, 0, stream>>>(psum, pcnt, l1W, l1b, l2W, l2b, out);
}